// DiffPooling_50182397886830
// MI455X (gfx1250) — compile-verified
//
#include <hip/hip_runtime.h>
#include <hip/hip_bf16.h>

typedef float v2f __attribute__((ext_vector_type(2)));
typedef float v8f __attribute__((ext_vector_type(8)));

#define IN_FEATS 256
#define KOUT 32
#define WPAD 260   // padded LDS row stride for transposed W (bank-conflict avoidance)

// ---------------- zero ----------------
__global__ void zero_f32(float* __restrict__ p, int n) {
    int i = blockIdx.x * blockDim.x + threadIdx.x;
    if (i < n) p[i] = 0.0f;
}

// ---------------- degree: deg[dst[e]] += 1 ----------------
__global__ void deg_kernel(const int* __restrict__ dst, float* __restrict__ deg, int nEdges) {
    int e = blockIdx.x * blockDim.x + threadIdx.x;
    if (e < nEdges) atomicAdd(&deg[dst[e]], 1.0f);
}

// ---------------- norm = rsqrt(max(deg,1)) (in place) ----------------
__global__ void norm_kernel(float* __restrict__ norm, int nNodes) {
    int i = blockIdx.x * blockDim.x + threadIdx.x;
    if (i < nNodes) {
        float d = norm[i];
        norm[i] = rsqrtf(fmaxf(d, 1.0f));
    }
}

// ---------------- h = (feat * norm) @ W via V_WMMA_F32_16X16X4_F32 ----------------
// Block = 256 threads = 8 waves; wave w owns rows [blk*128 + w*16, +16), both N-halves.
__global__ __launch_bounds__(256) void gemm_wmma_kernel(
    const float* __restrict__ feat, const float* __restrict__ W,
    const float* __restrict__ norm, float* __restrict__ h, int nNodes)
{
    __shared__ float Wt[KOUT * WPAD];   // W transposed: Wt[n][k]
    const int tid = threadIdx.x;

    // Stage W (256x32, row-major) transposed into LDS, coalesced global reads.
    for (int i = tid; i < IN_FEATS * KOUT; i += 256) {
        int k = i >> 5, n = i & 31;
        Wt[n * WPAD + k] = W[i];
    }
    __syncthreads();

    const int wave  = tid >> 5;
    const int lane  = tid & 31;
    const int lhalf = lane >> 4;   // 0: lanes 0-15, 1: lanes 16-31
    const int l16   = lane & 15;

    const int m0  = (blockIdx.x * 8 + wave) * 16;
    int rowA      = m0 + l16;
    int rowClamp  = rowA < nNodes ? rowA : (nNodes - 1);  // keep EXEC all-ones for WMMA
    const float na = norm[rowClamp];

    // A fragment source: lane L covers K = kk + lhalf*2 + {0,1} of its row.
    const float* featRow = feat + (size_t)rowClamp * IN_FEATS + lhalf * 2;
    // B fragment sources: N = l16 (tile 0) / 16+l16 (tile 1), K = kk + lhalf*2 + {0,1}.
    const float* bp0 = &Wt[l16 * WPAD + lhalf * 2];
    const float* bp1 = &Wt[(16 + l16) * WPAD + lhalf * 2];

    v8f c0 = {};
    v8f c1 = {};
    #pragma unroll 4
    for (int kk = 0; kk < IN_FEATS; kk += 4) {
        v2f a = *(const v2f*)(featRow + kk);
        a *= na;                                   // fold degree-norm into A
        v2f b0 = *(const v2f*)(bp0 + kk);
        v2f b1 = *(const v2f*)(bp1 + kk);
        // (neg_a, A, neg_b, B, c_mod, C, reuse_a, reuse_b)
        c0 = __builtin_amdgcn_wmma_f32_16x16x4_f32(false, a, false, b0, (short)0, c0, false, false);
        c1 = __builtin_amdgcn_wmma_f32_16x16x4_f32(false, a, false, b1, (short)0, c1, false, false);
    }

    // D layout: VGPR r -> row m0 + r + 8*lhalf, col l16 (c0) / 16+l16 (c1)
    float* hw = h + (size_t)(m0 + lhalf * 8) * KOUT + l16;
    if (m0 + 16 <= nNodes) {
        // fast path: whole tile in range, unguarded coalesced stores (scalar branch)
        #pragma unroll
        for (int r = 0; r < 8; ++r) {
            hw[r * KOUT]      = c0[r];
            hw[r * KOUT + 16] = c1[r];
        }
    } else {
        #pragma unroll
        for (int r = 0; r < 8; ++r) {
            int row = m0 + r + lhalf * 8;
            if (row < nNodes) {
                hw[r * KOUT]      = c0[r];
                hw[r * KOUT + 16] = c1[r];
            }
        }
    }
}

// ---------------- scatter: out[dst[e]] += h[src[e]] (one wave per edge, lane = feature) ----
__global__ __launch_bounds__(256) void scatter_kernel(
    const float* __restrict__ h, const int* __restrict__ src,
    const int* __restrict__ dst, float* __restrict__ out, int nEdges)
{
    int t = blockIdx.x * 256 + threadIdx.x;
    int f = t & 31;
    // edge index is wave-uniform: force it scalar so src/dst come via s_load
    int e = __builtin_amdgcn_readfirstlane(t >> 5);
    if (e < nEdges) {
        int s = src[e];
        int d = dst[e];
        float v = h[(size_t)s * KOUT + f];          // coalesced 128B row read
        atomicAdd(&out[(size_t)d * KOUT + f], v);   // global_atomic_add_f32, no return
    }
}

// ---------------- final scale: out[n] *= norm[n] ----------------
__global__ void scale_kernel(float* __restrict__ out, const float* __restrict__ norm, int nNodes) {
    int t = blockIdx.x * blockDim.x + threadIdx.x;
    int n = t >> 5;
    if (n < nNodes) out[t] *= norm[n];
}

extern "C" void kernel_launch(void* const* d_in, const int* in_sizes, int n_in,
                              void* d_out, int out_size, void* d_ws, size_t ws_size,
                              hipStream_t stream) {
    const float* feat = (const float*)d_in[0];
    const float* W    = (const float*)d_in[1];
    const int*   src  = (const int*)d_in[2];
    const int*   dst  = (const int*)d_in[3];
    float*       out  = (float*)d_out;

    const int nNodes = in_sizes[0] / IN_FEATS;   // 100000
    const int nEdges = in_sizes[2];              // 1600000

    // Workspace: [norm (nNodes f32)] [h (nNodes*32 f32)]
    float* norm = (float*)d_ws;
    float* h    = norm + (((size_t)nNodes + 255) & ~(size_t)255);

    const int B = 256;
    // 1) zero deg accumulator and output
    zero_f32<<<(nNodes + B - 1) / B, B, 0, stream>>>(norm, nNodes);
    zero_f32<<<(nNodes * KOUT + B - 1) / B, B, 0, stream>>>(out, nNodes * KOUT);
    // 2) degree
    deg_kernel<<<(nEdges + B - 1) / B, B, 0, stream>>>(dst, norm, nEdges);
    // 3) norm (in place)
    norm_kernel<<<(nNodes + B - 1) / B, B, 0, stream>>>(norm, nNodes);
    // 4) h = (feat*norm) @ W  -- WMMA, 128 rows per block
    gemm_wmma_kernel<<<(nNodes + 127) / 128, B, 0, stream>>>(feat, W, norm, h, nNodes);
    // 5) scatter-add over edges (one wave per edge)
    {
        long long threads = (long long)nEdges * KOUT;
        scatter_kernel<<<(unsigned)((threads + B - 1) / B), B, 0, stream>>>(h, src, dst, out, nEdges);
    }
    // 6) out *= norm
    scale_kernel<<<(nNodes * KOUT + B - 1) / B, B, 0, stream>>>(out, norm, nNodes);
}